// GraphSAGE_Pairwise_16183436771650
// MI455X (gfx1250) — compile-verified
//
#include <hip/hip_runtime.h>
#include <hip/hip_bf16.h>

#define NN   50000      // N_NODES
#define NE   800000     // N_EDGES
#define INC  64         // IN_CH
#define HID  96
#define NP   32768      // N_PAIRS
#define SLOPE 0.1f

typedef __attribute__((ext_vector_type(2))) float v2f;
typedef __attribute__((ext_vector_type(8))) float v8f;

// The builtin is only visible in the device pass; don't probe it on host.
#if defined(__HIP_DEVICE_COMPILE__)
#if !__has_builtin(__builtin_amdgcn_wmma_f32_16x16x4_f32)
#error "missing __builtin_amdgcn_wmma_f32_16x16x4_f32 on this toolchain"
#endif
#endif

__device__ __forceinline__ v8f wmma_f32(v2f a, v2f b, v8f c) {
  // D = A(16x4,f32) * B(4x16,f32) + C(16x16,f32)
  return __builtin_amdgcn_wmma_f32_16x16x4_f32(
      /*neg_a=*/false, a, /*neg_b=*/false, b,
      /*c_mod=*/(short)0, c, /*reuse_a=*/false, /*reuse_b=*/false);
}

__device__ __forceinline__ float lrelu(float v) { return v > 0.0f ? v : SLOPE * v; }

// ---------------------------------------------------------------------------
// Embedding: h[NN x HID] = x[NN x INC] @ W[INC x HID] + b
// Block: (32,6) = 6 waves; wave w computes the 16x16 tile at cols w*16.
// ---------------------------------------------------------------------------
__global__ __launch_bounds__(192)
void embed_kernel(const float* __restrict__ x, const float* __restrict__ W,
                  const float* __restrict__ b, float* __restrict__ h) {
  __shared__ float As[16][INC + 4];          // stride 68 -> conflict-free column reads
  const int lane = threadIdx.x, wv = threadIdx.y;
  const int t = wv * 32 + lane;
  const int m0 = blockIdx.x * 16;

  for (int i = t; i < 16 * INC; i += 192)
    As[i >> 6][i & 63] = x[(size_t)(m0 + (i >> 6)) * INC + (i & 63)];
  __syncthreads();

  const int row = lane & 15;
  const int kh  = (lane >> 4) << 1;          // lanes 16-31 hold K = k0+2,k0+3
  const int nc  = wv * 16 + row;             // output column for this lane
  const float bias = b[nc];
  v8f c;
  #pragma unroll
  for (int r = 0; r < 8; ++r) c[r] = bias;

  for (int k0 = 0; k0 < INC; k0 += 4) {
    v2f a, bb;
    a.x  = As[row][k0 + kh];
    a.y  = As[row][k0 + kh + 1];
    bb.x = W[(k0 + kh) * HID + nc];
    bb.y = W[(k0 + kh + 1) * HID + nc];
    c = wmma_f32(a, bb, c);
  }

  const int mrow = m0 + ((lane >> 4) << 3);  // lanes 16-31 write M = r+8
  #pragma unroll
  for (int r = 0; r < 8; ++r)
    h[(size_t)(mrow + r) * HID + nc] = c[r];
}

// ---------------------------------------------------------------------------
// Degree counts (layer-invariant, computed once).
// ---------------------------------------------------------------------------
__global__ __launch_bounds__(256)
void count_kernel(const int* __restrict__ dst, float* cnt) {
  int e = blockIdx.x * 256 + threadIdx.x;
  if (e < NE) unsafeAtomicAdd(&cnt[dst[e]], 1.0f);
}

// ---------------------------------------------------------------------------
// Edge scatter-add: agg[dst[e]] += h[src[e]]   (96 lanes per edge, 4 edges/blk)
// ---------------------------------------------------------------------------
__global__ __launch_bounds__(384)
void scatter_kernel(const float* __restrict__ h, const int* __restrict__ src,
                    const int* __restrict__ dst, float* agg) {
  const int e = blockIdx.x * 4 + threadIdx.y;
  const int c = threadIdx.x;                 // 0..95
  const int s = src[e], d = dst[e];
  unsafeAtomicAdd(&agg[(size_t)d * HID + c], h[(size_t)s * HID + c]);
}

// ---------------------------------------------------------------------------
// SAGE combine: out = (agg/max(cnt,1)) @ Wl + bl + h @ Wr  [+ LeakyReLU]
// In-place over agg is safe: each block stages its 16 rows in LDS first.
// ---------------------------------------------------------------------------
__global__ __launch_bounds__(192)
void sage_kernel(const float* agg, const float* __restrict__ h,
                 const float* __restrict__ Wl, const float* __restrict__ bl,
                 const float* __restrict__ Wr, const float* __restrict__ cnt,
                 float* out, int leaky) {
  __shared__ float Sa[16][HID + 4];          // stride 100 -> conflict-free
  __shared__ float Sh[16][HID + 4];
  __shared__ float Sinv[16];
  const int lane = threadIdx.x, wv = threadIdx.y;
  const int t = wv * 32 + lane;
  const int m0 = blockIdx.x * 16;

  for (int i = t; i < 16 * HID; i += 192) {
    int r = i / HID, cc = i - r * HID;
    Sa[r][cc] = agg[(size_t)(m0 + r) * HID + cc];
    Sh[r][cc] = h[(size_t)(m0 + r) * HID + cc];
  }
  if (t < 16) Sinv[t] = 1.0f / fmaxf(cnt[m0 + t], 1.0f);
  __syncthreads();

  const int row = lane & 15;
  const int kh  = (lane >> 4) << 1;
  const int nc  = wv * 16 + row;
  const float inv = Sinv[row];
  const float bias = bl[nc];
  v8f c;
  #pragma unroll
  for (int r = 0; r < 8; ++r) c[r] = bias;

  for (int k0 = 0; k0 < HID; k0 += 4) {      // mean-neighbor branch
    v2f a, bb;
    a.x  = Sa[row][k0 + kh] * inv;
    a.y  = Sa[row][k0 + kh + 1] * inv;
    bb.x = Wl[(k0 + kh) * HID + nc];
    bb.y = Wl[(k0 + kh + 1) * HID + nc];
    c = wmma_f32(a, bb, c);
  }
  for (int k0 = 0; k0 < HID; k0 += 4) {      // root branch
    v2f a, bb;
    a.x  = Sh[row][k0 + kh];
    a.y  = Sh[row][k0 + kh + 1];
    bb.x = Wr[(k0 + kh) * HID + nc];
    bb.y = Wr[(k0 + kh + 1) * HID + nc];
    c = wmma_f32(a, bb, c);
  }

  const int mrow = m0 + ((lane >> 4) << 3);
  #pragma unroll
  for (int r = 0; r < 8; ++r) {
    float v = c[r];
    if (leaky) v = lrelu(v);
    out[(size_t)(mrow + r) * HID + nc] = v;
  }
}

// ---------------------------------------------------------------------------
// Pair MLP layer 1: ph = leaky(concat(h[a],h[b]) @ W1[192x96] + b1)
// ---------------------------------------------------------------------------
__global__ __launch_bounds__(192)
void pair_mlp1_kernel(const float* __restrict__ h, const int* __restrict__ pairs,
                      const float* __restrict__ W1, const float* __restrict__ b1,
                      float* __restrict__ ph) {
  __shared__ float Sp[16][2 * HID + 4];      // stride 196 -> conflict-free
  __shared__ int Pi[32];
  const int lane = threadIdx.x, wv = threadIdx.y;
  const int t = wv * 32 + lane;
  const int p0 = blockIdx.x * 16;

  if (t < 32) Pi[t] = pairs[p0 * 2 + t];
  __syncthreads();
  for (int i = t; i < 16 * 2 * HID; i += 192) {
    int r = i / (2 * HID), cc = i - r * 2 * HID;
    int node = Pi[2 * r + (cc >= HID ? 1 : 0)];
    Sp[r][cc] = h[(size_t)node * HID + (cc >= HID ? cc - HID : cc)];
  }
  __syncthreads();

  const int row = lane & 15;
  const int kh  = (lane >> 4) << 1;
  const int nc  = wv * 16 + row;
  const float bias = b1[nc];
  v8f c;
  #pragma unroll
  for (int r = 0; r < 8; ++r) c[r] = bias;

  for (int k0 = 0; k0 < 2 * HID; k0 += 4) {
    v2f a, bb;
    a.x  = Sp[row][k0 + kh];
    a.y  = Sp[row][k0 + kh + 1];
    bb.x = W1[(k0 + kh) * HID + nc];
    bb.y = W1[(k0 + kh + 1) * HID + nc];
    c = wmma_f32(a, bb, c);
  }

  const int prow = p0 + ((lane >> 4) << 3);
  #pragma unroll
  for (int r = 0; r < 8; ++r)
    ph[(size_t)(prow + r) * HID + nc] = lrelu(c[r]);
}

// ---------------------------------------------------------------------------
// Output head: out[p] = ph[p] . W2 + b2   (one wave per pair, shfl reduction)
// ---------------------------------------------------------------------------
__global__ __launch_bounds__(256)
void out_kernel(const float* __restrict__ ph, const float* __restrict__ W2,
                const float* __restrict__ b2, float* __restrict__ out) {
  const int p = blockIdx.x * 8 + threadIdx.y;
  const int lane = threadIdx.x;
  float s = 0.0f;
  #pragma unroll
  for (int k = lane; k < HID; k += 32)
    s += ph[(size_t)p * HID + k] * W2[k];
  #pragma unroll
  for (int off = 16; off > 0; off >>= 1)
    s += __shfl_down(s, off, 32);
  if (lane == 0) out[p] = s + b2[0];
}

// ---------------------------------------------------------------------------
extern "C" void kernel_launch(void* const* d_in, const int* in_sizes, int n_in,
                              void* d_out, int out_size, void* d_ws, size_t ws_size,
                              hipStream_t stream) {
  (void)in_sizes; (void)n_in; (void)out_size; (void)ws_size;
  const float* x    = (const float*)d_in[0];
  const int*   edge = (const int*)d_in[1];    // [2, NE]
  const int*   prs  = (const int*)d_in[2];    // [NP, 2]
  const float* Wemb = (const float*)d_in[3];
  const float* bemb = (const float*)d_in[4];
  const float* Wl   = (const float*)d_in[5];  // [3,96,96]
  const float* bl   = (const float*)d_in[6];  // [3,96]
  const float* Wr   = (const float*)d_in[7];  // [3,96,96]
  const float* W1   = (const float*)d_in[8];  // [192,96]
  const float* b1   = (const float*)d_in[9];
  const float* W2   = (const float*)d_in[10]; // [96,1]
  const float* b2   = (const float*)d_in[11];
  float* out = (float*)d_out;

  const int* src = edge;
  const int* dst = edge + NE;

  float* bufA = (float*)d_ws;                       // h features (ping)
  float* bufB = bufA + (size_t)NN * HID;            // agg / h features (pong)
  float* cnt  = bufB + (size_t)NN * HID;            // degree counts
  float* ph   = cnt + NN;                           // pair hidden [NP x HID]

  const dim3 blkG(32, 6);

  (void)hipMemsetAsync(cnt, 0, NN * sizeof(float), stream);
  count_kernel<<<(NE + 255) / 256, 256, 0, stream>>>(dst, cnt);
  embed_kernel<<<NN / 16, blkG, 0, stream>>>(x, Wemb, bemb, bufA);

  float* hcur = bufA;
  float* hoth = bufB;
  for (int l = 0; l < 3; ++l) {
    (void)hipMemsetAsync(hoth, 0, (size_t)NN * HID * sizeof(float), stream);
    scatter_kernel<<<NE / 4, dim3(96, 4), 0, stream>>>(hcur, src, dst, hoth);
    sage_kernel<<<NN / 16, blkG, 0, stream>>>(
        hoth, hcur, Wl + (size_t)l * HID * HID, bl + (size_t)l * HID,
        Wr + (size_t)l * HID * HID, cnt, hoth, (l < 2) ? 1 : 0);
    float* tmp = hcur; hcur = hoth; hoth = tmp;
  }

  pair_mlp1_kernel<<<NP / 16, blkG, 0, stream>>>(hcur, prs, W1, b1, ph);
  out_kernel<<<NP / 8, dim3(32, 8), 0, stream>>>(ph, W2, b2, out);
}